// RNN_63247688400977
// MI455X (gfx1250) — compile-verified
//
#include <hip/hip_runtime.h>
#include <hip/hip_bf16.h>
#include <math.h>

typedef __attribute__((ext_vector_type(2))) float v2f;
typedef __attribute__((ext_vector_type(8))) float v8f;
typedef __attribute__((ext_vector_type(4))) int   v4i;

#define SEQ 8192
#define IN  2048
#define HID 2048

// CDNA5 async global->LDS copy (GLOBAL_LOAD_ASYNC_TO_LDS_B128, ASYNCcnt).
// Builtin signature (from hipcc diagnostic): (v4i* global, v4i* lds, Ii, Ii).
__device__ __forceinline__ void async_b128(void* lds, const void* g) {
  __builtin_amdgcn_global_load_async_to_lds_b128(
      (__attribute__((address_space(1))) v4i*)(v4i*)(void*)(g),
      (__attribute__((address_space(3))) v4i*)(v4i*)(lds),
      /*imm offset*/ 0, /*cpol*/ 0);
}

// ===================== Phase 1: x_proj = x @ Wi^T + bi =====================
// fp32 WMMA 16x16x4. Block = 256 threads = 8 waves arranged 2(M) x 4(N),
// computing a 32x64 output tile. K chunked by 32 through double-buffered LDS,
// filled by async-to-LDS DMA overlapped with the WMMA pipe.
__global__ __launch_bounds__(256) void xproj_wmma(const float* __restrict__ x,
                                                  const float* __restrict__ Wi,
                                                  const float* __restrict__ bi,
                                                  float* __restrict__ out) {
  __shared__ __align__(16) float xs[2][32][36];  // x tile  [M=32][K=32] (+pad, 144B rows)
  __shared__ __align__(16) float ws[2][64][36];  // Wi tile [N=64][K=32] (+pad)
  const int tid  = threadIdx.x;
  const int wave = tid >> 5;
  const int lane = tid & 31;
  const int half = lane >> 4;   // 0: lanes 0-15, 1: lanes 16-31
  const int lm   = lane & 15;
  const int wm   = wave & 1;    // m-tile within block (0..1)
  const int wn   = wave >> 1;   // n-tile within block (0..3)
  const int tileN = blockIdx.x * 64;
  const int tileM = blockIdx.y * 32;

  // One b128 per thread for xs (32 rows x 8 quads), two for ws (64 x 8).
  auto load_chunk = [&](int buf, int k0) {
    {
      const int r = tid >> 3, q = tid & 7;
      async_b128(&xs[buf][r][q * 4],
                 x + (size_t)(tileM + r) * IN + k0 + q * 4);
    }
#pragma unroll
    for (int j = 0; j < 2; ++j) {
      const int i = tid + j * 256;
      const int r = i >> 3, q = i & 7;
      async_b128(&ws[buf][r][q * 4],
                 Wi + (size_t)(tileN + r) * IN + k0 + q * 4);
    }
  };

  v8f c = {};
  load_chunk(0, 0);
  __builtin_amdgcn_s_wait_asynccnt(0);
  __syncthreads();

  int buf = 0;
  for (int k0 = 0; k0 < IN; k0 += 32, buf ^= 1) {
    if (k0 + 32 < IN) load_chunk(buf ^ 1, k0 + 32);  // prefetch next chunk
#pragma unroll
    for (int kk = 0; kk < 32; kk += 4) {
      // A frag (16x4 f32): lanes 0-15 -> K=kk..kk+1, lanes 16-31 -> K=kk+2..+3
      v2f a, b;
      a.x = xs[buf][wm * 16 + lm][kk + half * 2 + 0];
      a.y = xs[buf][wm * 16 + lm][kk + half * 2 + 1];
      // B frag (4x16 f32), mirrored; lane lm carries column n=lm of B,
      // i.e. row (tileN + wn*16 + lm) of Wi.
      b.x = ws[buf][wn * 16 + lm][kk + half * 2 + 0];
      b.y = ws[buf][wn * 16 + lm][kk + half * 2 + 1];
      c = __builtin_amdgcn_wmma_f32_16x16x4_f32(false, a, false, b, (short)0, c,
                                                false, false);
    }
    __builtin_amdgcn_s_wait_asynccnt(0);
    __syncthreads();
  }

  // C/D layout: VGPR r -> M = r (lanes 0-15) / r+8 (lanes 16-31); N = lm.
  const int gn = tileN + wn * 16 + lm;
  const float bias = bi[gn];
#pragma unroll
  for (int r = 0; r < 8; ++r) {
    const int gm = tileM + wm * 16 + r + half * 8;
    out[(size_t)gm * HID + gn] = c[r] + bias;
  }
}

// ===================== Phase 2: sequential tanh recurrence =====================
#define NWG 64
#define ROWS_PER_WG (HID / NWG)  // 32 rows -> 32*2048*4 = 256 KB of Wh in LDS

__global__ __launch_bounds__(256) void rnn_init(float* h0, unsigned* bar) {
  const int i = blockIdx.x * 256 + threadIdx.x;
  if (i < HID) h0[i] = 0.0f;
  if (i < 2) bar[i] = 0u;
}

__global__ __launch_bounds__(256) void rnn_scan(const float* __restrict__ Wh,
                                                const float* __restrict__ bh,
                                                float* __restrict__ out,   // [SEQ,HID]: x_proj in, h out (in place)
                                                float* __restrict__ hn,    // [HID] final state
                                                float* __restrict__ hbuf0,
                                                float* __restrict__ hbuf1,
                                                unsigned* __restrict__ bar) {
  extern __shared__ float lds[];
  float* whl = lds;                          // ROWS_PER_WG x HID slice of Wh
  float* hl  = lds + ROWS_PER_WG * HID;      // h vector (HID)
  const int tid  = threadIdx.x;
  const int wave = tid >> 5;
  const int lane = tid & 31;
  const int row0 = blockIdx.x * ROWS_PER_WG;

  // Pin this WG's Wh slice in LDS for the whole scan (CDNA5: 320 KB/WGP),
  // streamed in with async-to-LDS DMA (64 b128 per thread, one wait).
  for (int i = tid * 4; i < ROWS_PER_WG * HID; i += 256 * 4) {
    async_b128(whl + i, Wh + (size_t)row0 * HID + i);
  }
  __builtin_amdgcn_s_wait_asynccnt(0);
  __syncthreads();

  // Hoist loop-invariant bias loads (4 rows per wave).
  float bh_r[ROWS_PER_WG / 8];
#pragma unroll
  for (int rr = 0; rr < ROWS_PER_WG / 8; ++rr) {
    bh_r[rr] = bh[row0 + wave * (ROWS_PER_WG / 8) + rr];
  }

  unsigned* cnt = bar;
  unsigned* gen = bar + 1;

  for (int t = 0; t < SEQ; ++t) {
    const float* hin  = (t & 1) ? hbuf1 : hbuf0;
    float*       hout = (t & 1) ? hbuf0 : hbuf1;
    // Async-copy h_{t-1} (8 KB) into LDS: 2 b128 per thread.
    for (int i = tid * 4; i < HID; i += 256 * 4) {
      async_b128(hl + i, hin + i);
    }
    __builtin_amdgcn_s_wait_asynccnt(0);
    __syncthreads();

    // 8 waves x 4 rows each = 32 rows; lane-strided dot, wave32 xor-reduce.
#pragma unroll
    for (int rr = 0; rr < ROWS_PER_WG / 8; ++rr) {
      const int rl = wave * (ROWS_PER_WG / 8) + rr;
      const float* wrow = whl + rl * HID;
      float acc = 0.0f;
      for (int k = lane; k < HID; k += 32) acc = fmaf(wrow[k], hl[k], acc);
      for (int off = 16; off > 0; off >>= 1) acc += __shfl_xor(acc, off, 32);
      if (lane == 0) {
        const int g = row0 + rl;
        const float v = tanhf(out[(size_t)t * HID + g] + acc + bh_r[rr]);
        out[(size_t)t * HID + g] = v;
        hout[g] = v;
        if (t == SEQ - 1) hn[g] = v;
      }
    }
    __syncthreads();

    // Device-wide generation barrier (deterministic, capture-safe).
    if (tid == 0) {
      const unsigned arrived = __hip_atomic_fetch_add(
          cnt, 1u, __ATOMIC_ACQ_REL, __HIP_MEMORY_SCOPE_AGENT);
      if (arrived == NWG - 1) {
        __hip_atomic_store(cnt, 0u, __ATOMIC_RELAXED, __HIP_MEMORY_SCOPE_AGENT);
        __hip_atomic_store(gen, (unsigned)(t + 1), __ATOMIC_RELEASE,
                           __HIP_MEMORY_SCOPE_AGENT);
      } else {
        while (__hip_atomic_load(gen, __ATOMIC_ACQUIRE,
                                 __HIP_MEMORY_SCOPE_AGENT) < (unsigned)(t + 1)) {
          __builtin_amdgcn_s_sleep(2);
        }
      }
    }
    __syncthreads();
  }
}

extern "C" void kernel_launch(void* const* d_in, const int* in_sizes, int n_in,
                              void* d_out, int out_size, void* d_ws, size_t ws_size,
                              hipStream_t stream) {
  const float* x  = (const float*)d_in[0];
  const float* Wi = (const float*)d_in[1];
  const float* bi = (const float*)d_in[2];
  const float* Wh = (const float*)d_in[3];
  const float* bh = (const float*)d_in[4];
  float* out = (float*)d_out;                 // outputs [SEQ,HID]
  float* hn  = out + (size_t)SEQ * HID;       // h_n [HID]

  float* hbuf0 = (float*)d_ws;
  float* hbuf1 = hbuf0 + HID;
  unsigned* bar = (unsigned*)(hbuf1 + HID);

  rnn_init<<<(HID + 255) / 256, 256, 0, stream>>>(hbuf0, bar);

  dim3 g1(HID / 64, SEQ / 32);
  xproj_wmma<<<g1, 256, 0, stream>>>(x, Wi, bi, out);

  const size_t ldsBytes = (size_t)(ROWS_PER_WG * HID + HID) * sizeof(float);
  rnn_scan<<<NWG, 256, ldsBytes, stream>>>(Wh, bh, out, hn, hbuf0, hbuf1, bar);
}